// EAConv_77738908057844
// MI455X (gfx1250) — compile-verified
//
#include <hip/hip_runtime.h>
#include <math.h>

typedef float v2f __attribute__((ext_vector_type(2)));
typedef float v8f __attribute__((ext_vector_type(8)));

// ---------------- Kernel A: Wcat = [sum_l Wl | W1] (64x128), blSum = sum_l bl ----
__global__ void ea_prep(const float* __restrict__ Wl, const float* __restrict__ bl,
                        const float* __restrict__ W1, float* __restrict__ Wcat,
                        float* __restrict__ blSum) {
  int i = blockIdx.x * blockDim.x + threadIdx.x;
  if (i < 64 * 128) {
    int kk = i >> 7, c = i & 127;
    float v;
    if (c < 64) {
      v = 0.f;
#pragma unroll
      for (int l = 0; l < 6; ++l) v += Wl[l * 4096 + kk * 64 + c];
    } else {
      v = W1[kk * 64 + (c - 64)];
    }
    Wcat[i] = v;
  } else if (i < 64 * 128 + 64) {
    int c = i - 8192;
    float v = 0.f;
#pragma unroll
    for (int l = 0; l < 6; ++l) v += bl[l * 64 + c];
    blSum[c] = v;
  }
}

// ---------------- Kernel B: xw = x @ WlSum ; xw1 = x @ W1  (WMMA f32 16x16x4) ----
// Block: 256 thr = 8 waves. 16 rows x 128 cols per block; wave w -> col tile w.
__global__ __launch_bounds__(256) void ea_gemm_x(
    const float* __restrict__ x, const float* __restrict__ Wcat,
    float* __restrict__ xw, float* __restrict__ xw1) {
  __shared__ float lx[16][68];    // padded: column reads conflict-free
  __shared__ float lw[64][136];   // padded: split-half row reads conflict-free
  int tid = threadIdx.x;
  int row0 = blockIdx.x * 16;
  for (int i = tid; i < 16 * 64; i += 256)
    lx[i >> 6][i & 63] = x[(row0 + (i >> 6)) * 64 + (i & 63)];
  for (int i = tid; i < 64 * 128; i += 256)
    lw[i >> 7][i & 127] = Wcat[i];
  __syncthreads();

  int wave = tid >> 5, lane = tid & 31;
  int l = lane & 15, hi = lane >> 4;
  int col0 = wave * 16;
  v8f acc = {};
#pragma unroll
  for (int kk = 0; kk < 16; ++kk) {
    v2f a, b;
    a.x = lx[l][kk * 4 + 2 * hi];            // A 16x4: lanes<16 K={0,1}, lanes>=16 K={2,3}
    a.y = lx[l][kk * 4 + 2 * hi + 1];
    b.x = lw[kk * 4 + 2 * hi][col0 + l];     // B 4x16: same split
    b.y = lw[kk * 4 + 2 * hi + 1][col0 + l];
    acc = __builtin_amdgcn_wmma_f32_16x16x4_f32(false, a, false, b, (short)0, acc,
                                                false, false);
  }
  float* dst = (wave < 4) ? xw : xw1;
  int c = ((wave & 3) << 4) + l;
#pragma unroll
  for (int g = 0; g < 8; ++g) {              // D: VGPR g -> row g (lanes<16) / g+8
    int r = row0 + g + hi * 8;
    dst[r * 64 + c] = acc[g];
  }
}

// ---------------- Kernel C1: per-node geometry -> coef[t][j], gp[t] ----
__global__ void ea_geom(const float* __restrict__ p, const int* __restrict__ sid,
                        const float* __restrict__ Wm, const float* __restrict__ bm,
                        float* __restrict__ coef, float* __restrict__ gp, int N) {
  int t = blockIdx.x * blockDim.x + threadIdx.x;
  if (t >= N) return;
  float ptx = p[t * 3 + 0], pty = p[t * 3 + 1], ptz = p[t * 3 + 2];
  float w0 = Wm[0], w1 = Wm[1], w2 = Wm[2], w3 = Wm[3];
  float base = ptx * w0 + pty * w1 + ptz * w2 + bm[0];
  float dis[16], ang[16];
  float gmax = -3.4e38f, dmax = 0.f;
#pragma unroll
  for (int j = 0; j < 16; ++j) {
    int s = sid[t * 16 + j];
    float dx = p[s * 3 + 0] - ptx;
    float dy = p[s * 3 + 1] - pty;
    float dz = p[s * 3 + 2] - ptz;
    float dn = fmaxf(sqrtf(dx * dx + dy * dy + dz * dz), 1e-16f);
    float xy = fmaxf(sqrtf(dx * dx + dy * dy), 1e-16f);
    dis[j] = dn;
    ang[j] = cosf(dz / dn) * cosf(dx / xy);
    gmax = fmaxf(gmax, base + dn * w3);
    dmax = fmaxf(dmax, dn);
  }
  float pr = dmax * 1.1f;
  float pd[16], sum = 0.f;
#pragma unroll
  for (int j = 0; j < 16; ++j) {
    float d = pr - dis[j];
    pd[j] = d * d;
    sum += pd[j];
  }
  float inv = 1.f / sum;
#pragma unroll
  for (int j = 0; j < 16; ++j) coef[t * 16 + j] = ang[j] * pd[j] * inv;
  gp[t] = gmax;
}

// ---------------- Kernel C2: y[t] = sum_j c_j*xw[s_j] + (sum c_j)*(blSum - xw[t])
// + per-block BN1 partial sums. Block: 256 thr = 4 node-slots x 64 channels; 16 nodes.
__global__ __launch_bounds__(256) void ea_agg(
    const float* __restrict__ xw, const float* __restrict__ coef,
    const int* __restrict__ sid, const float* __restrict__ blSum,
    float* __restrict__ y_raw, float* __restrict__ pSum, float* __restrict__ pSq) {
  __shared__ float scoef[4][16];
  __shared__ int ssid[4][16];
  __shared__ float red[4][64], red2[4][64];
  int tid = threadIdx.x;
  int c = tid & 63, q = tid >> 6;
  int node0 = blockIdx.x * 16;
  float bls = blSum[c];
  float ls = 0.f, ls2 = 0.f;
  for (int it = 0; it < 4; ++it) {
    int t = node0 + it * 4 + q;
    __syncthreads();
    if (c < 16) {
      scoef[q][c] = coef[t * 16 + c];
      ssid[q][c] = sid[t * 16 + c];
    }
    __syncthreads();
    float acc = 0.f, sc = 0.f;
#pragma unroll
    for (int j = 0; j < 16; ++j) {
      float cj = scoef[q][j];
      acc = fmaf(cj, xw[ssid[q][j] * 64 + c], acc);  // coalesced L2-resident gather
      sc += cj;
    }
    float y = fmaf(sc, bls - xw[t * 64 + c], acc);
    y_raw[t * 64 + c] = y;
    ls += y;
    ls2 += y * y;
  }
  red[q][c] = ls;
  red2[q][c] = ls2;
  __syncthreads();
  if (q == 0) {  // deterministic fixed-order reduce
    pSum[blockIdx.x * 64 + c] = red[0][c] + red[1][c] + red[2][c] + red[3][c];
    pSq[blockIdx.x * 64 + c] = red2[0][c] + red2[1][c] + red2[2][c] + red2[3][c];
  }
}

// ---------------- BN reduce: AB[c]=gamma*rsqrt(var+eps); AB[64+c]=beta-mean*AB[c]
__global__ void ea_bnred(const float* __restrict__ pSum, const float* __restrict__ pSq,
                         int nblocks, float Ninv, const float* __restrict__ g,
                         const float* __restrict__ be, float* __restrict__ AB) {
  int c = threadIdx.x;
  if (c >= 64) return;
  float s = 0.f, s2 = 0.f;
  for (int b = 0; b < nblocks; ++b) {
    s += pSum[b * 64 + c];
    s2 += pSq[b * 64 + c];
  }
  float mean = s * Ninv;
  float var = s2 * Ninv - mean * mean;
  float A = g[c] * rsqrtf(var + 1e-5f);
  AB[c] = A;
  AB[64 + c] = be[c] - mean * A;
}

// ---------------- Kernel E: z = relu(bn1(y))@W2 + gp*xw1 + b1 + b2 (WMMA) + BN2 partials
// Block: 256 thr = 8 waves = 2 row-tiles x 4 col-tiles; 32 rows per block.
__global__ __launch_bounds__(256) void ea_gemm_y(
    const float* __restrict__ y_raw, const float* __restrict__ W2,
    const float* __restrict__ AB1, const float* __restrict__ gp,
    const float* __restrict__ xw1, const float* __restrict__ b1,
    const float* __restrict__ b2, float* __restrict__ z,
    float* __restrict__ pSum, float* __restrict__ pSq) {
  __shared__ float ly[32][68];
  __shared__ float lw[64][72];
  __shared__ float lred[2][64], lred2[2][64];
  int tid = threadIdx.x;
  int row0 = blockIdx.x * 32;
  for (int i = tid; i < 32 * 64; i += 256) {
    int r = i >> 6, c = i & 63;
    float v = y_raw[(row0 + r) * 64 + c];
    ly[r][c] = fmaxf(fmaf(AB1[c], v, AB1[64 + c]), 0.f);  // fused BN1 + ReLU
  }
  for (int i = tid; i < 64 * 64; i += 256) lw[i >> 6][i & 63] = W2[i];
  __syncthreads();

  int wave = tid >> 5, lane = tid & 31;
  int l = lane & 15, hi = lane >> 4;
  int m = wave >> 2, nt = wave & 3;
  int col0 = nt * 16;
  v8f acc = {};
#pragma unroll
  for (int kk = 0; kk < 16; ++kk) {
    v2f a, b;
    a.x = ly[m * 16 + l][kk * 4 + 2 * hi];
    a.y = ly[m * 16 + l][kk * 4 + 2 * hi + 1];
    b.x = lw[kk * 4 + 2 * hi][col0 + l];
    b.y = lw[kk * 4 + 2 * hi + 1][col0 + l];
    acc = __builtin_amdgcn_wmma_f32_16x16x4_f32(false, a, false, b, (short)0, acc,
                                                false, false);
  }
  int c = col0 + l;
  float bias = b1[c] + b2[c];
  float s = 0.f, s2 = 0.f;
#pragma unroll
  for (int g = 0; g < 8; ++g) {
    int r = row0 + m * 16 + g + hi * 8;
    float zz = acc[g] + gp[r] * xw1[r * 64 + c] + bias;  // residual branch
    z[r * 64 + c] = zz;
    s += zz;
    s2 += zz * zz;
  }
  s += __shfl_xor(s, 16, 32);    // lanes L and L+16 share a column
  s2 += __shfl_xor(s2, 16, 32);
  if (hi == 0) {
    lred[m][c] = s;
    lred2[m][c] = s2;
  }
  __syncthreads();
  if (tid < 64) {
    pSum[blockIdx.x * 64 + tid] = lred[0][tid] + lred[1][tid];
    pSq[blockIdx.x * 64 + tid] = lred2[0][tid] + lred2[1][tid];
  }
}

// ---------------- Final: out = relu(bn2(z)) ----
__global__ void ea_final(const float* __restrict__ z, const float* __restrict__ AB2,
                         float* __restrict__ out, int total) {
  int i = blockIdx.x * blockDim.x + threadIdx.x;
  if (i >= total) return;
  int c = i & 63;
  out[i] = fmaxf(fmaf(AB2[c], z[i], AB2[64 + c]), 0.f);
}

extern "C" void kernel_launch(void* const* d_in, const int* in_sizes, int n_in,
                              void* d_out, int out_size, void* d_ws, size_t ws_size,
                              hipStream_t stream) {
  const float* x   = (const float*)d_in[0];
  const float* p   = (const float*)d_in[1];
  const int*   sid = (const int*)d_in[4];
  const float* Wl  = (const float*)d_in[6];
  const float* bl  = (const float*)d_in[7];
  const float* W1  = (const float*)d_in[8];
  const float* b1  = (const float*)d_in[9];
  const float* W2  = (const float*)d_in[10];
  const float* b2  = (const float*)d_in[11];
  const float* Wm  = (const float*)d_in[12];
  const float* bm  = (const float*)d_in[13];
  const float* g1  = (const float*)d_in[14];
  const float* be1 = (const float*)d_in[15];
  const float* g2  = (const float*)d_in[16];
  const float* be2 = (const float*)d_in[17];

  const int N = in_sizes[0] / 64;  // 65536

  // workspace carve-out (256B aligned)
  char* base = (char*)d_ws;
  size_t off = 0;
  auto take = [&](size_t bytes) {
    char* ptr = base + off;
    off = (off + bytes + 255) & ~(size_t)255;
    return ptr;
  };
  float* Wcat  = (float*)take(64 * 128 * sizeof(float));
  float* blSum = (float*)take(64 * sizeof(float));
  float* AB1   = (float*)take(128 * sizeof(float));
  float* AB2   = (float*)take(128 * sizeof(float));
  float* xw    = (float*)take((size_t)N * 64 * sizeof(float));  // reused as z
  float* xw1   = (float*)take((size_t)N * 64 * sizeof(float));
  float* coef  = (float*)take((size_t)N * 16 * sizeof(float));
  float* gp    = (float*)take((size_t)N * sizeof(float));
  float* y_raw = (float*)take((size_t)N * 64 * sizeof(float));
  int nblkC2 = N / 16;
  int nblkE  = N / 32;
  float* p1s = (float*)take((size_t)nblkC2 * 64 * sizeof(float));
  float* p1q = (float*)take((size_t)nblkC2 * 64 * sizeof(float));
  float* p2s = (float*)take((size_t)nblkE * 64 * sizeof(float));
  float* p2q = (float*)take((size_t)nblkE * 64 * sizeof(float));
  float* zb  = xw;  // xw is dead after ea_agg; alias for z

  float Ninv = 1.f / (float)N;

  ea_prep<<<(64 * 128 + 64 + 255) / 256, 256, 0, stream>>>(Wl, bl, W1, Wcat, blSum);
  ea_gemm_x<<<N / 16, 256, 0, stream>>>(x, Wcat, xw, xw1);
  ea_geom<<<(N + 255) / 256, 256, 0, stream>>>(p, sid, Wm, bm, coef, gp, N);
  ea_agg<<<nblkC2, 256, 0, stream>>>(xw, coef, sid, blSum, y_raw, p1s, p1q);
  ea_bnred<<<1, 64, 0, stream>>>(p1s, p1q, nblkC2, Ninv, g1, be1, AB1);
  ea_gemm_y<<<nblkE, 256, 0, stream>>>(y_raw, W2, AB1, gp, xw1, b1, b2, zb, p2s, p2q);
  ea_bnred<<<1, 64, 0, stream>>>(p2s, p2q, nblkE, Ninv, g2, be2, AB2);
  ea_final<<<(N * 64 + 255) / 256, 256, 0, stream>>>(zb, AB2, (float*)d_out, N * 64);
}